// FrustumPooling_721554506291
// MI455X (gfx1250) — compile-verified
//
#include <hip/hip_runtime.h>

typedef __attribute__((ext_vector_type(2))) float v2f;
typedef __attribute__((ext_vector_type(8))) float v8f;

#define B_  2
#define N_  4
#define D_  48
#define H_  28
#define W_  60
#define C_  64
#define NX_ 192
#define NY_ 192
// f_bev = 1/SCALE = 4, cx = NX/2 + OFFSETX = 96, cy = NY/2 = 96

__device__ __forceinline__ float rlane_f(float v, int l) {
  return __builtin_bit_cast(float, __builtin_amdgcn_readlane(__builtin_bit_cast(int, v), l));
}

__global__ __launch_bounds__(256) void zero4_kernel(float4* __restrict__ p, int n4) {
  int i = blockIdx.x * 256 + threadIdx.x;
  if (i < n4) p[i] = make_float4(0.f, 0.f, 0.f, 0.f);
}

// One wave (32 lanes) transforms 16 frustum points with a single
// V_WMMA_F32_16X16X4_F32 (A = 16x4 homogeneous pts, B = 4x16 carrying the
// 3 rows of [R*inv(K) | t]), extracts x/y/z per point via v_readlane, and
// scatter-adds the point's 64 channels (2 per lane) with global f32 atomics.
// mode 0: acc laid out (b, y, x, c)  -> contiguous 256B atomic burst per point
// mode 1: acc is the final output (b, c, y, x) -> channel stride NY*NX
__global__ __launch_bounds__(256) void frustum_scatter(
    const float* __restrict__ x, const float* __restrict__ intr,
    const float* __restrict__ pose, float* __restrict__ acc, int mode)
{
  const int lane = threadIdx.x & 31;
  const int wave = threadIdx.x >> 5;
  const int p0   = (blockIdx.x * 8 + wave) * 16;   // 16 points per wave
  const int m    = lane & 15;
  const int p    = p0 + m;

  // decompose point id -> (b, n, d, h, w); 16 consecutive points share (b,n)
  int w  = p % W_;  int t = p / W_;
  int h  = t % H_;  t /= H_;
  int dd = t % D_;  t /= D_;
  int n  = t % N_;
  int b  = t / N_;

  const float dval = 2.0f + (float)dd;                 // DMIN + dd*DSTEP
  const float u    = (float)w * (479.0f / 59.0f);      // linspace(0, 479, 60)
  const float v    = (float)h * (223.0f / 27.0f);      // linspace(0, 223, 28)

  const bool hi = lane >= 16;
  // A 16x4 f32 layout: lanes 0-15 hold (K0,K1), lanes 16-31 hold (K2,K3), M = lane%16
  v2f A;
  A.x = hi ? dval : u * dval;
  A.y = hi ? 1.0f : v * dval;

  // --- per-(b,n) transform: M = pose_rot * inv(K), wave-uniform ---
  const float* Kp = intr + (size_t)(b * N_ + n) * 9;
  float k00 = Kp[0], k01 = Kp[1], k02 = Kp[2];
  float k10 = Kp[3], k11 = Kp[4], k12 = Kp[5];
  float k20 = Kp[6], k21 = Kp[7], k22 = Kp[8];
  float det = k00 * (k11 * k22 - k12 * k21)
            - k01 * (k10 * k22 - k12 * k20)
            + k02 * (k10 * k21 - k11 * k20);
  float id  = 1.0f / det;
  float i00 =  (k11 * k22 - k12 * k21) * id;
  float i01 = -(k01 * k22 - k02 * k21) * id;
  float i02 =  (k01 * k12 - k02 * k11) * id;
  float i10 = -(k10 * k22 - k12 * k20) * id;
  float i11 =  (k00 * k22 - k02 * k20) * id;
  float i12 = -(k00 * k12 - k02 * k10) * id;
  float i20 =  (k10 * k21 - k11 * k20) * id;
  float i21 = -(k00 * k21 - k01 * k20) * id;
  float i22 =  (k00 * k11 - k01 * k10) * id;

  const float* Pp = pose + (size_t)(b * N_ + n) * 16;
  float r00 = Pp[0], r01 = Pp[1], r02 = Pp[2],  tx = Pp[3];
  float r10 = Pp[4], r11 = Pp[5], r12 = Pp[6],  ty = Pp[7];
  float r20 = Pp[8], r21 = Pp[9], r22 = Pp[10], tz = Pp[11];

  float M00 = r00 * i00 + r01 * i10 + r02 * i20;
  float M01 = r00 * i01 + r01 * i11 + r02 * i21;
  float M02 = r00 * i02 + r01 * i12 + r02 * i22;
  float M10 = r10 * i00 + r11 * i10 + r12 * i20;
  float M11 = r10 * i01 + r11 * i11 + r12 * i21;
  float M12 = r10 * i02 + r11 * i12 + r12 * i22;
  float M20 = r20 * i00 + r21 * i10 + r22 * i20;
  float M21 = r20 * i01 + r21 * i11 + r22 * i21;
  float M22 = r20 * i02 + r21 * i12 + r22 * i22;

  // B 4x16 f32 layout: lane%16 = output column n; lanes 0-15 hold (K0,K1),
  // lanes 16-31 hold (K2,K3). Column n<3 is row n of [M | t], columns 3..15 zero.
  const int nc = lane & 15;
  float v0 = nc == 0 ? M00 : nc == 1 ? M10 : nc == 2 ? M20 : 0.0f;  // k=0
  float v1 = nc == 0 ? M01 : nc == 1 ? M11 : nc == 2 ? M21 : 0.0f;  // k=1
  float v2 = nc == 0 ? M02 : nc == 1 ? M12 : nc == 2 ? M22 : 0.0f;  // k=2
  float v3 = nc == 0 ? tx  : nc == 1 ? ty  : nc == 2 ? tz  : 0.0f;  // k=3
  v2f Bv;
  Bv.x = hi ? v2 : v0;
  Bv.y = hi ? v3 : v1;

  v8f Dm = {0.f, 0.f, 0.f, 0.f, 0.f, 0.f, 0.f, 0.f};
  Dm = __builtin_amdgcn_wmma_f32_16x16x4_f32(false, A, false, Bv,
                                             (short)0, Dm, false, false);

  const long cs = (mode == 0) ? 1L : (long)(NY_ * NX_);
  const float* xrow = x + (long)p0 * C_ + lane * 2;   // this point-block's rows

  // D 16x16 f32 layout: vgpr r holds row r (lanes 0-15) and row r+8 (lanes 16-31)
#pragma unroll
  for (int mm = 0; mm < 16; ++mm) {
    float dr = (mm < 8) ? Dm[mm] : Dm[mm - 8];
    const int sl = (mm < 8) ? 0 : 16;
    float gx_f = rlane_f(dr, sl + 0) * 4.0f + 96.0f;
    float gy_f = rlane_f(dr, sl + 1) * 4.0f + 96.0f;
    float gz_f = (rlane_f(dr, sl + 2) + 10.0f) * (1.0f / 20.0f);  // mul, not IEEE div chain
    int gx = (int)gx_f;            // trunc-toward-zero matches astype(int32)
    int gy = (int)gy_f;
    int gz = (int)gz_f;
    if (gx >= 0 && gx < NX_ && gy >= 0 && gy < NY_ && gz == 0) {
      long base;
      if (mode == 0) base = ((long)(b * NY_ + gy) * NX_ + gx) * (long)C_;
      else           base = (long)b * C_ * (NY_ * NX_) + (long)gy * NX_ + gx;
      const float2 xv = *(const float2*)(xrow + (long)mm * C_);  // one global_load_b64
      atomicAdd(acc + base + (long)(lane * 2) * cs,      xv.x);
      atomicAdd(acc + base + (long)(lane * 2) * cs + cs, xv.y);
    }
  }
}

// ws (b, y, x, c) -> out (b, c, y, x) via a 64x64 LDS tile (padded to 65 to
// stay conflict-free on 64 banks). Coalesced on both sides.
__global__ __launch_bounds__(256) void bev_transpose(const float* __restrict__ ws,
                                                     float* __restrict__ out)
{
  __shared__ float tile[64][65];
  int tb = blockIdx.x;
  int xt = tb % (NX_ / 64);  tb /= (NX_ / 64);
  int y  = tb % NY_;
  int b  = tb / NY_;

  const int tid = threadIdx.x;
  {
    int c = tid & 63, q = tid >> 6;
    const float* src = ws + (((long)(b * NY_ + y)) * NX_ + xt * 64) * (long)C_;
#pragma unroll
    for (int i = 0; i < 16; ++i) {
      int xs = q * 16 + i;
      tile[xs][c] = src[(long)xs * C_ + c];
    }
  }
  __syncthreads();
  {
    int xs = tid & 63, cq = tid >> 6;
#pragma unroll
    for (int j = 0; j < 16; ++j) {
      int c = cq * 16 + j;
      out[((long)(b * C_ + c) * NY_ + y) * NX_ + xt * 64 + xs] = tile[xs][c];
    }
  }
}

extern "C" void kernel_launch(void* const* d_in, const int* in_sizes, int n_in,
                              void* d_out, int out_size, void* d_ws, size_t ws_size,
                              hipStream_t stream) {
  const float* x    = (const float*)d_in[0];
  const float* intr = (const float*)d_in[1];
  const float* pose = (const float*)d_in[2];
  float* out = (float*)d_out;

  const int nOut = B_ * C_ * NY_ * NX_;            // 4,718,592 floats
  const size_t wsNeed = (size_t)nOut * sizeof(float);
  const int nPoints = B_ * N_ * D_ * H_ * W_;      // 645,120
  const int scatterBlocks = nPoints / 128;         // 16 pts/wave * 8 waves/block
  const int zeroBlocks = (nOut / 4 + 255) / 256;

  if (d_ws && ws_size >= wsNeed) {
    float* acc = (float*)d_ws;
    zero4_kernel<<<zeroBlocks, 256, 0, stream>>>((float4*)acc, nOut / 4);
    frustum_scatter<<<scatterBlocks, 256, 0, stream>>>(x, intr, pose, acc, 0);
    bev_transpose<<<B_ * NY_ * (NX_ / 64), 256, 0, stream>>>(acc, out);
  } else {
    zero4_kernel<<<zeroBlocks, 256, 0, stream>>>((float4*)out, nOut / 4);
    frustum_scatter<<<scatterBlocks, 256, 0, stream>>>(x, intr, pose, out, 1);
  }
}